// AdaptiveGraphAttention_2783138807993
// MI455X (gfx1250) — compile-verified
//
#include <hip/hip_runtime.h>

// AdaptiveGraphAttention — MI455X (gfx1250, wave32) optimized.
//
// Key exact algebra: softmax is invariant to per-row constants, and the mask
// (adj) only kills column j==0 identically for all rows i. Hence
// attn[b,h,i,:] is independent of i, ctx rows are identical, and the whole
// layer reduces to one attention vector per (b,h) plus one output row per b,
// broadcast over the 256 sequence positions. Wq/bq and attn_b are dead.
//
// Memory-bound plan (~100 MB traffic ~ 4-5us @ 23.3 TB/s):
//   K1: U[h,:] = sum_d w_k[d] * Wk[h*64+d,:]           (16x1024, trivial)
//   K2: g[b,h,j] = nv[b,j,:].U[h] + desc[b,j-1,h].w_e + bias   (j=0 -> NEG)
//   K3: softmax_j(g) in place -> attn
//   K4: m[b] = attn_b(16x256) @ nv_b(256x1024)   <-- V_WMMA_F32_16X16X4_F32
//   K5: ctx[b,h*64+d] = m[b,h,:].Wv[h*64+d,:] + bv
//   K6: row[b,o]      = ctx[b,:].Wo[o,:] + bo
//   K7: out[b,i,:] = row[b,:]  (float4 coalesced broadcast)

#define NEG (-1000000000.0f)

typedef __attribute__((ext_vector_type(2))) float v2f;
typedef __attribute__((ext_vector_type(8))) float v8f;

constexpr int B = 32, S = 256, D = 1024, H = 16, HD = 64;

__device__ __forceinline__ float wave_reduce_add(float v) {
#pragma unroll
  for (int off = 16; off; off >>= 1) v += __shfl_xor(v, off, 32);
  return v;
}

// K1: fold Wk through w_k: U[h][c], and abias[h] = sum_d bk[h*64+d]*w_k[d]
__global__ void k1_foldK(const float* __restrict__ Wk, const float* __restrict__ bk,
                         const float* __restrict__ attn_w,
                         float* __restrict__ U, float* __restrict__ abias) {
  int idx = blockIdx.x * blockDim.x + threadIdx.x;  // 16*1024 threads
  int h = idx >> 10, c = idx & 1023;
  float acc = 0.f;
#pragma unroll 8
  for (int d = 0; d < HD; ++d)
    acc += attn_w[HD + d] * Wk[(size_t)(h * HD + d) * D + c];
  U[idx] = acc;
  if (c == 0) {
    float ab = 0.f;
    for (int d = 0; d < HD; ++d) ab += bk[h * HD + d] * attn_w[HD + d];
    abias[h] = ab;
  }
}

// K2: masked pre-softmax logits. Block = one (b,j); 16 waves = 16 heads.
__global__ void k2_logits(const float* __restrict__ nv, const float* __restrict__ desc,
                          const float* __restrict__ attn_w,
                          const float* __restrict__ U, const float* __restrict__ abias,
                          float* __restrict__ g) {
  __shared__ float row[D];
  int b = blockIdx.x >> 8, j = blockIdx.x & 255;
  int t = threadIdx.x;
  const float* nvrow = nv + (size_t)(b * S + j) * D;
  for (int i = t; i < D; i += blockDim.x) row[i] = nvrow[i];
  __syncthreads();
  int h = t >> 5, lane = t & 31;
  float part = 0.f;
#pragma unroll 4
  for (int c = lane; c < D; c += 32) part += row[c] * U[h * D + c];
  if (j > 0) {  // e_j term (padded row 0 is masked anyway)
    const float* drow = desc + (size_t)(b * (S - 1) + (j - 1)) * D + h * HD;
    part += drow[lane] * attn_w[2 * HD + lane]
          + drow[lane + 32] * attn_w[2 * HD + lane + 32];
  }
  part = wave_reduce_add(part);
  if (lane == 0)
    g[(size_t)(b * H + h) * S + j] = (j == 0) ? NEG : (part + abias[h]);
}

// K3: softmax over j, in place. Block = one (b,h) row of 256.
__global__ void k3_softmax(float* __restrict__ g) {
  __shared__ float sd[S];
  int t = threadIdx.x;
  float* gp = g + (size_t)blockIdx.x * S;
  float x = gp[t];
  sd[t] = x; __syncthreads();
  for (int s2 = 128; s2 > 0; s2 >>= 1) {
    if (t < s2) sd[t] = fmaxf(sd[t], sd[t + s2]);
    __syncthreads();
  }
  float mx = sd[0]; __syncthreads();
  float e = __expf(x - mx);        // j==0 underflows to exactly 0
  sd[t] = e; __syncthreads();
  for (int s2 = 128; s2 > 0; s2 >>= 1) {
    if (t < s2) sd[t] += sd[t + s2];
    __syncthreads();
  }
  gp[t] = e / sd[0];
}

// K4: m[b](16x1024) = attn[b](16x256) @ nv[b](256x1024) via f32 WMMA.
// One wave per (b, 16-col N tile); 64 K-steps of 16x16x4.
// A 16x4 layout (ISA 7.12.2): lane l holds row M=l&15; VGPR0/1 = K pair
// {2*(l>>4), 2*(l>>4)+1}. B 4x16 assumed symmetric; C/D per 16x16 f32 table.
__global__ void k4_mix_wmma(const float* __restrict__ attn, const float* __restrict__ nv,
                            float* __restrict__ m) {
  int wid = (blockIdx.x * blockDim.x + threadIdx.x) >> 5;
  int lane = threadIdx.x & 31;
  int b = wid >> 6;           // 64 N-tiles per batch
  int nt = wid & 63;
  int mrow = lane & 15;       // M index for A, N index for B/D
  int khalf = lane >> 4;
  int n = nt * 16 + mrow;
  const float* A = attn + (size_t)b * H * S;   // [16][256]
  const float* Bm = nv + (size_t)b * S * D;    // [256][1024]
  v8f acc = {0.f, 0.f, 0.f, 0.f, 0.f, 0.f, 0.f, 0.f};
#pragma unroll 4
  for (int k0 = 0; k0 < S; k0 += 4) {
    int k = k0 + 2 * khalf;
    v2f a, bb;
    a.x = A[mrow * S + k];
    a.y = A[mrow * S + k + 1];
    bb.x = Bm[(size_t)k * D + n];
    bb.y = Bm[(size_t)(k + 1) * D + n];
    acc = __builtin_amdgcn_wmma_f32_16x16x4_f32(
        /*neg_a=*/false, a, /*neg_b=*/false, bb,
        /*c_mod=*/(short)0, acc, /*reuse_a=*/false, /*reuse_b=*/false);
  }
  float* mp = m + (size_t)b * H * D + nt * 16 + mrow;
#pragma unroll
  for (int r = 0; r < 8; ++r)
    mp[(size_t)(r + 8 * khalf) * D] = acc[r];
}

// K5: ctx[b, r=h*64+d] = m[b,h,:].Wv[r,:] + bv[r]. One wave per output.
__global__ void k5_ctx(const float* __restrict__ m, const float* __restrict__ Wv,
                       const float* __restrict__ bv, float* __restrict__ ctx) {
  int gw = (blockIdx.x * blockDim.x + threadIdx.x) >> 5;
  int lane = threadIdx.x & 31;
  int b = gw >> 10, r = gw & 1023, h = r >> 6;
  const float* mp = m + (size_t)(b * H + h) * D;
  const float* wp = Wv + (size_t)r * D;
  float acc = 0.f;
#pragma unroll 4
  for (int c = lane; c < D; c += 32) acc += mp[c] * wp[c];
  acc = wave_reduce_add(acc);
  if (lane == 0) ctx[b * D + r] = acc + bv[r];
}

// K6: row[b,o] = ctx[b,:].Wo[o,:] + bo[o]. One wave per output.
__global__ void k6_out_row(const float* __restrict__ ctx, const float* __restrict__ Wo,
                           const float* __restrict__ bo, float* __restrict__ rowv) {
  int gw = (blockIdx.x * blockDim.x + threadIdx.x) >> 5;
  int lane = threadIdx.x & 31;
  int b = gw >> 10, o = gw & 1023;
  const float* cp = ctx + (size_t)b * D;
  const float* wp = Wo + (size_t)o * D;
  float acc = 0.f;
#pragma unroll 4
  for (int c = lane; c < D; c += 32) acc += cp[c] * wp[c];
  acc = wave_reduce_add(acc);
  if (lane == 0) rowv[b * D + o] = acc + bo[o];
}

// K7: out[b,i,:] = row[b,:], float4-vectorized (33.5 MB streaming stores).
__global__ void k7_bcast(const float4* __restrict__ rowv, float4* __restrict__ out) {
  int idx = blockIdx.x * blockDim.x + threadIdx.x;  // 2,097,152 float4s
  int b = idx >> 16;        // 65536 float4s per batch
  int q = idx & 255;        // float4 column within the 1024-wide row
  out[idx] = rowv[(b << 8) + q];
}

extern "C" void kernel_launch(void* const* d_in, const int* in_sizes, int n_in,
                              void* d_out, int out_size, void* d_ws, size_t ws_size,
                              hipStream_t stream) {
  (void)in_sizes; (void)n_in; (void)out_size; (void)ws_size;
  const float* desc   = (const float*)d_in[0];
  const float* nv     = (const float*)d_in[1];
  // d_in[2]=Wq, d_in[3]=bq: dead (a_q is a per-row softmax constant)
  const float* Wk     = (const float*)d_in[4];
  const float* bk     = (const float*)d_in[5];
  const float* Wv     = (const float*)d_in[6];
  const float* bv     = (const float*)d_in[7];
  const float* attn_w = (const float*)d_in[8];
  // d_in[9]=attn_b: dead (uniform logit shift cancels in softmax)
  const float* Wo     = (const float*)d_in[10];
  const float* bo     = (const float*)d_in[11];
  float* out = (float*)d_out;

  float* ws    = (float*)d_ws;          // ~2.95 MB total
  float* U     = ws;                    // 16*1024        = 16384
  float* abias = ws + 16384;            // 16
  float* g     = ws + 16448;            // 32*16*256      = 131072 (logits->attn)
  float* m     = ws + 147520;           // 32*16*1024     = 524288
  float* ctx   = ws + 671808;           // 32*1024        = 32768
  float* rowv  = ws + 704576;           // 32*1024        = 32768 (16B aligned)

  k1_foldK   <<<64,    256, 0, stream>>>(Wk, bk, attn_w, U, abias);
  k2_logits  <<<B * S, 512, 0, stream>>>(nv, desc, attn_w, U, abias, g);
  k3_softmax <<<B * H, S,   0, stream>>>(g);
  k4_mix_wmma<<<256,   256, 0, stream>>>(g, nv, m);       // 2048 waves
  k5_ctx     <<<4096,  256, 0, stream>>>(m, Wv, bv, ctx); // 32768 waves
  k6_out_row <<<4096,  256, 0, stream>>>(ctx, Wo, bo, rowv);
  k7_bcast   <<<8192,  256, 0, stream>>>((const float4*)rowv, (float4*)out);
}